// TemporalIntegrator_35304631173378
// MI455X (gfx1250) — compile-verified
//
#include <hip/hip_runtime.h>

// ---------------------------------------------------------------------------
// TemporalIntegrator for MI455X (gfx1250, wave32).
//
// v = v0 + cumsum(dv),  dv = dt/6*(a_i + 4*a_mid + a_n)
// z = z0 + cumsum(dz),  dz = dt/6*(6*v_prev + dt*(a_i + 2*a_mid))
//
// Memory-bound (~335 MB compulsory => ~14us @ 23.3 TB/s). Strategy:
// 3-level scan, run twice (v then z), 256-element chunk-local scans done
// with V_WMMA_F32_16X16X4_F32 (fp32-exact triangular-matmul scan, 8 WMMA
// per 256-element scan per wave). Re-reads of accel/t between passes hit
// the 192MB L2 (inputs are 134MB total), so HBM traffic stays ~compulsory.
// Row-sum broadcasts use v_readlane (SGPR, no LDS round-trip) instead of
// ds_bpermute.
// ---------------------------------------------------------------------------

typedef float v2f __attribute__((ext_vector_type(2)));
typedef float v8f __attribute__((ext_vector_type(8)));

#define TS      256                 // timesteps per chunk (= blockDim.x)
#define ROWPAD  18                  // LDS row stride (even => b64-aligned, bank-conflict free)
#define TILE_F  (16 * ROWPAD)       // floats per component tile
#define LAST_IDX (15 * ROWPAD + 15) // location of element 255 after scan

__device__ __forceinline__ int lpos(int i) { return (i >> 4) * ROWPAD + (i & 15); }

// Broadcast lane `src` (compile-time 15 or 31) of x to all lanes via v_readlane.
__device__ __forceinline__ float bcast_lane(float x, int src) {
  return __int_as_float(__builtin_amdgcn_readlane(__float_as_int(x), src));
}

// Wave32 in-place inclusive scan of 256 f32 stored in a padded 16x18 LDS tile.
// Element p lives at s[(p>>4)*ROWPAD + (p&15)]  (row-major 16x16 + pad).
// Step 1: D = A * U   (U upper-triangular ones)  -> row-wise cumsums
// Step 2: D += Lstrict * broadcast(rowsums)      -> add cross-row offsets
// Both via chained V_WMMA_F32_16X16X4_F32 (K=4 x 4). fp32-exact.
__device__ __forceinline__ void wmma_scan256(float* s) {
  const int lane = threadIdx.x & 31;
  const int m = lane & 15;   // A row
  const int h = lane >> 4;   // half-wave select
  const int n = m;           // B/C/D column for this lane

  v8f d = {};
  // ---- group 1: row-wise inclusive scan -------------------------------
#pragma unroll
  for (int k = 0; k < 4; ++k) {
    const int col = 4 * k + 2 * h;
    v2f a, b;
    a[0] = s[ROWPAD * m + col];
    a[1] = s[ROWPAD * m + col + 1];
    b[0] = ((col + 0) <= n) ? 1.0f : 0.0f;   // U[k][n] = (k<=n)
    b[1] = ((col + 1) <= n) ? 1.0f : 0.0f;
    d = __builtin_amdgcn_wmma_f32_16x16x4_f32(false, a, false, b,
                                              (short)0, d, false, false);
  }
  // ---- broadcast the 16 row sums (column 15 of D) via v_readlane ------
  float rs[16];
#pragma unroll
  for (int r = 0; r < 8; ++r) {
    rs[r]     = bcast_lane(d[r], 15);  // rows 0..7
    rs[r + 8] = bcast_lane(d[r], 31);  // rows 8..15
  }
  // ---- group 2: D += Lstrict * B2,  B2[k][n] = rowsum[k] --------------
#pragma unroll
  for (int k = 0; k < 4; ++k) {
    const int col = 4 * k + 2 * h;
    v2f a, b;
    a[0] = ((col + 0) < m) ? 1.0f : 0.0f;    // Lstrict[m][k] = (k<m)
    a[1] = ((col + 1) < m) ? 1.0f : 0.0f;
    b[0] = h ? rs[4 * k + 2] : rs[4 * k + 0];
    b[1] = h ? rs[4 * k + 3] : rs[4 * k + 1];
    d = __builtin_amdgcn_wmma_f32_16x16x4_f32(false, a, false, b,
                                              (short)0, d, false, false);
  }
  // ---- store back (C/D layout: lane holds D[r+8h][n] in vgpr r) -------
#pragma unroll
  for (int r = 0; r < 8; ++r) s[ROWPAD * (r + 8 * h) + n] = d[r];
}

// dv for timestep p (valid p < N-1)
__device__ __forceinline__ void load_dv(const float* __restrict__ accel,
                                        const float* __restrict__ t,
                                        long tb, float& dt,
                                        float& dvx, float& dvy, float& dvz,
                                        float& tx, float& ty, float& tz) {
  dt = t[tb + 1] - t[tb];
  const float* a0 = accel + tb * 3;
  const float aix = a0[0], aiy = a0[1], aiz = a0[2];
  const float anx = a0[3], any_ = a0[4], anz = a0[5];
  const float amx = 0.5f * (aix + anx);
  const float amy = 0.5f * (aiy + any_);
  const float amz = 0.5f * (aiz + anz);
  const float f = dt * (1.0f / 6.0f);
  dvx = f * (aix + 4.0f * amx + anx);
  dvy = f * (aiy + 4.0f * amy + any_);
  dvz = f * (aiz + 4.0f * amz + anz);
  tx = aix + 2.0f * amx;   // a_i + 2*a_mid  (for dz)
  ty = aiy + 2.0f * amy;
  tz = aiz + 2.0f * amz;
}

// ---- pass 1: per-chunk sums of dv ----------------------------------------
__global__ __launch_bounds__(TS)
void k_chunk_sums_v(const float* __restrict__ accel, const float* __restrict__ t,
                    float* __restrict__ Sv, int N, int C) {
  const int c = blockIdx.x, b = blockIdx.y, i = threadIdx.x;
  const long p = (long)c * TS + i;
  __shared__ float sc[3][TILE_F];
  float dt, dvx = 0.f, dvy = 0.f, dvz = 0.f, tx, ty, tz;
  if (p < N - 1) load_dv(accel, t, (long)b * N + p, dt, dvx, dvy, dvz, tx, ty, tz);
  const int q = lpos(i);
  sc[0][q] = dvx; sc[1][q] = dvy; sc[2][q] = dvz;
  __syncthreads();
  const int wid = i >> 5;
  if (wid < 3) wmma_scan256(sc[wid]);
  __syncthreads();
  if (i < 3) Sv[((long)b * C + c) * 3 + i] = sc[i][LAST_IDX];
}

// ---- pass 2/4: per-batch exclusive scan of chunk sums (+ initial value) ---
__global__ __launch_bounds__(32)
void k_offset_scan(const float* __restrict__ S, const float* __restrict__ init,
                   float* __restrict__ O, int C) {
  const int b = blockIdx.x, lane = threadIdx.x;
  float cx = init[b * 3 + 0], cy = init[b * 3 + 1], cz = init[b * 3 + 2];
  for (int c0 = 0; c0 < C; c0 += 32) {
    const int c = c0 + lane;
    const long base = ((long)b * C + c) * 3;
    float x = 0.f, y = 0.f, z = 0.f;
    if (c < C) { x = S[base]; y = S[base + 1]; z = S[base + 2]; }
    float ix = x, iy = y, iz = z;
#pragma unroll
    for (int dlt = 1; dlt < 32; dlt <<= 1) {
      float ox = __shfl_up(ix, dlt, 32);
      float oy = __shfl_up(iy, dlt, 32);
      float oz = __shfl_up(iz, dlt, 32);
      if (lane >= dlt) { ix += ox; iy += oy; iz += oz; }
    }
    if (c < C) {
      O[base]     = cx + ix - x;   // exclusive
      O[base + 1] = cy + iy - y;
      O[base + 2] = cz + iz - z;
    }
    cx += bcast_lane(ix, 31);
    cy += bcast_lane(iy, 31);
    cz += bcast_lane(iz, 31);
  }
}

// ---- pass 3: write v, produce per-chunk sums of dz ------------------------
__global__ __launch_bounds__(TS)
void k_write_v_sums_z(const float* __restrict__ accel, const float* __restrict__ t,
                      const float* __restrict__ v0, const float* __restrict__ z0,
                      const float* __restrict__ Ov,
                      float* __restrict__ vout, float* __restrict__ zout,
                      float* __restrict__ Sz, int N, int C) {
  const int c = blockIdx.x, b = blockIdx.y, i = threadIdx.x;
  const long p = (long)c * TS + i;
  __shared__ float sc[3][TILE_F];
  const bool valid = (p < N - 1);
  float dt = 0.f, dvx = 0.f, dvy = 0.f, dvz = 0.f, tx = 0.f, ty = 0.f, tz = 0.f;
  if (valid) load_dv(accel, t, (long)b * N + p, dt, dvx, dvy, dvz, tx, ty, tz);
  const int q = lpos(i);
  sc[0][q] = dvx; sc[1][q] = dvy; sc[2][q] = dvz;
  __syncthreads();
  const int wid = i >> 5;
  if (wid < 3) wmma_scan256(sc[wid]);
  __syncthreads();

  const long ob = ((long)b * C + c) * 3;
  const float ovx = Ov[ob], ovy = Ov[ob + 1], ovz = Ov[ob + 2];
  float dzx = 0.f, dzy = 0.f, dzz = 0.f;
  if (valid) {
    const float vix = ovx + sc[0][q];     // v[p+1]
    const float viy = ovy + sc[1][q];
    const float viz = ovz + sc[2][q];
    const long vb = ((long)b * N + p + 1) * 3;
    vout[vb] = vix; vout[vb + 1] = viy; vout[vb + 2] = viz;
    const float vpx = vix - dvx, vpy = viy - dvy, vpz = viz - dvz;  // v_prev
    const float g = dt * (1.0f / 6.0f);
    dzx = g * (6.0f * vpx + dt * tx);
    dzy = g * (6.0f * vpy + dt * ty);
    dzz = g * (6.0f * vpz + dt * tz);
  }
  if (c == 0 && i == 0) {                 // v[0]=v0, z[0]=z0
    const long vb = (long)b * N * 3;
    vout[vb] = v0[b * 3]; vout[vb + 1] = v0[b * 3 + 1]; vout[vb + 2] = v0[b * 3 + 2];
    zout[vb] = z0[b * 3]; zout[vb + 1] = z0[b * 3 + 1]; zout[vb + 2] = z0[b * 3 + 2];
  }
  // each LDS slot is re-written only by its owner thread after the sync above
  sc[0][q] = dzx; sc[1][q] = dzy; sc[2][q] = dzz;
  __syncthreads();
  if (wid < 3) wmma_scan256(sc[wid]);
  __syncthreads();
  if (i < 3) Sz[ob + i] = sc[i][LAST_IDX];
}

// ---- pass 5: write z (v_prev read back from d_out, L2-resident) -----------
__global__ __launch_bounds__(TS)
void k_write_z(const float* __restrict__ accel, const float* __restrict__ t,
               const float* __restrict__ vout, const float* __restrict__ Oz,
               float* __restrict__ zout, int N, int C) {
  const int c = blockIdx.x, b = blockIdx.y, i = threadIdx.x;
  const long p = (long)c * TS + i;
  __shared__ float sc[3][TILE_F];
  const bool valid = (p < N - 1);
  float dzx = 0.f, dzy = 0.f, dzz = 0.f;
  if (valid) {
    float dt, dvx, dvy, dvz, tx, ty, tz;
    const long tb = (long)b * N + p;
    load_dv(accel, t, tb, dt, dvx, dvy, dvz, tx, ty, tz);
    const float* vp = vout + tb * 3;      // v_prev = v[p]
    const float g = dt * (1.0f / 6.0f);
    dzx = g * (6.0f * vp[0] + dt * tx);
    dzy = g * (6.0f * vp[1] + dt * ty);
    dzz = g * (6.0f * vp[2] + dt * tz);
  }
  const int q = lpos(i);
  sc[0][q] = dzx; sc[1][q] = dzy; sc[2][q] = dzz;
  __syncthreads();
  const int wid = i >> 5;
  if (wid < 3) wmma_scan256(sc[wid]);
  __syncthreads();
  if (valid) {
    const long ob = ((long)b * C + c) * 3;
    const long zb = ((long)b * N + p + 1) * 3;
    zout[zb]     = Oz[ob]     + sc[0][q];
    zout[zb + 1] = Oz[ob + 1] + sc[1][q];
    zout[zb + 2] = Oz[ob + 2] + sc[2][q];
  }
}

extern "C" void kernel_launch(void* const* d_in, const int* in_sizes, int n_in,
                              void* d_out, int out_size, void* d_ws, size_t ws_size,
                              hipStream_t stream) {
  const float* accel = (const float*)d_in[0];
  const float* v0    = (const float*)d_in[1];
  const float* z0    = (const float*)d_in[2];
  const float* t     = (const float*)d_in[3];

  const int B = in_sizes[1] / 3;          // v0 is (B,3)
  const int N = in_sizes[3] / B;          // t is (B,N,1)
  const int C = (N - 2) / TS + 1;         // ceil((N-1)/TS)

  float* vout = (float*)d_out;
  float* zout = vout + (size_t)B * N * 3;

  const size_t BC3 = (size_t)B * C * 3;
  float* Sv = (float*)d_ws;               // chunk sums of dv
  float* Ov = Sv + BC3;                   // chunk-start v offsets
  float* Sz = Ov + BC3;                   // chunk sums of dz
  float* Oz = Sz + BC3;                   // chunk-start z offsets

  dim3 grid(C, B);
  k_chunk_sums_v  <<<grid, TS, 0, stream>>>(accel, t, Sv, N, C);
  k_offset_scan   <<<B,    32, 0, stream>>>(Sv, v0, Ov, C);
  k_write_v_sums_z<<<grid, TS, 0, stream>>>(accel, t, v0, z0, Ov, vout, zout, Sz, N, C);
  k_offset_scan   <<<B,    32, 0, stream>>>(Sz, z0, Oz, C);
  k_write_z       <<<grid, TS, 0, stream>>>(accel, t, vout, Oz, zout, N, C);
}